// SinusoidalSynthesizer_111669150212
// MI455X (gfx1250) — compile-verified
//
#include <hip/hip_runtime.h>
#include <math.h>

// ---------------------------------------------------------------------------
// SinusoidalSynthesizer (DDSP-style) for MI455X / gfx1250.
//
// out0: amplitude_envelopes [8, 64000, 100]  (51.2M fp32)
// out1: frequency_envelopes [8,   250, 100]  (0.2M fp32, == freqs)
//
// Store-bandwidth bound: 205.6 MB / 23.3 TB/s ~ 9us floor. The overlap-add
// upsample reduces to a per-sample crossfade of two frame rows:
//   out[b,n,c] = amps[b,s,c]*w1[h] + amps[b,s-1,c]*(1-w1[h]),
//   s = n/256 + 1, h = n%256, w1[h] = 0.5*(1-cos(pi*h/256)).
// We express each 16-sample x 16-channel tile as a 16x4 @ 4x16 fp32 matmul
// via V_WMMA_F32_16X16X4_F32 (exact in fp32; K=2,3 padded with zeros).
// ---------------------------------------------------------------------------

typedef __attribute__((ext_vector_type(2))) float v2f;
typedef __attribute__((ext_vector_type(8))) float v8f;

#define NB   8
#define NF   250
#define NC   100
#define NS   64000
#define HOP  256
#define LOG10F      2.302585092994046f
#define MIDI_MAXF   119.21309485364912f   // 12*(log2(8000)-log2(440))+69
#define PI_OVER_256 0.0122718463030851f   // pi/256

// ---- Kernel 1: get_controls (elementwise, 200K elems) ----------------------
__global__ __launch_bounds__(256)
void controls_kernel(const float* __restrict__ amp_in,
                     const float* __restrict__ frq_in,
                     float* __restrict__ amps_ws,     // [8,250,100] masked amps
                     float* __restrict__ freq_out) {  // d_out tail
    int idx = blockIdx.x * 256 + threadIdx.x;
    if (idx >= NB * NF * NC) return;
    int c = idx % NC;

    float a = amp_in[idx];
    float f = frq_in[idx];

    // amps = 2*sigmoid(a)^ln10 + 1e-7
    float sa   = 1.0f / (1.0f + __expf(-a));
    float amps = 2.0f * __powf(sa, LOG10F) + 1e-7f;

    // freqs = 440 * 2^((sigmoid(f)*MIDI_MAX - 69)/12)
    float unit = 1.0f / (1.0f + __expf(-f));
    float midi = unit * MIDI_MAXF;
    // 2^x = exp(x*ln2); ln2/12 = 0.057762265...
    float freq = 440.0f * __expf((midi - 69.0f) * 0.05776226504666211f);

    // remove_above_nyquist: aa = (freq*(c+1) < sr/2) + 1e-4
    float aa = ((freq * (float)(c + 1) < 22050.0f) ? 1.0f : 0.0f) + 1e-4f;

    amps_ws[idx]  = amps * aa;
    freq_out[idx] = freq;
}

// ---- Kernel 2: windowed upsample via WMMA f32 16x16x4 ----------------------
// One workgroup (256 thr = 8 waves) per (batch b, slot s): 256 samples x 100 ch.
__global__ __launch_bounds__(256)
void upsample_wmma_kernel(const float* __restrict__ amps_ws,
                          float* __restrict__ out) {
    __shared__ float a_cur[112];   // amps row s   (zero-padded 100 -> 112)
    __shared__ float a_prv[112];   // amps row s-1
    __shared__ float w1[HOP];      // first-half hann window

    const int tid = threadIdx.x;
    const int blk = blockIdx.x;
    const int b   = blk / NF;
    const int sm1 = blk % NF;                 // slot s = sm1+1, prev frame = sm1
    const int row_s = min(sm1 + 1, NF - 1);   // duplicated last frame

    if (tid < 112) {
        a_cur[tid] = (tid < NC) ? amps_ws[(b * NF + row_s) * NC + tid] : 0.0f;
    } else if (tid < 224) {
        int j = tid - 112;
        a_prv[j] = (j < NC) ? amps_ws[(b * NF + sm1) * NC + j] : 0.0f;
    }
    // periodic hann, first half: w1[h] = 0.5*(1 - cos(2*pi*h/512))
    w1[tid] = 0.5f - 0.5f * __cosf((float)tid * PI_OVER_256);
    __syncthreads();

    const int wave = tid >> 5;
    const int lane = tid & 31;
    const int col  = lane & 15;
    const bool lo  = (lane < 16);

    const int ob_base = b * (NS * NC) + sm1 * (HOP * NC);

    // 16 sample-tiles x 7 channel-tiles = 112 tiles; 14 per wave.
    #pragma unroll 1
    for (int i = 0; i < 14; ++i) {
        const int t  = wave * 14 + i;
        const int m0 = (t / 7) << 4;     // sample-tile origin (phase h)
        const int c0 = (t % 7) << 4;     // channel-tile origin

        // A (16x4, MxK): lanes 0-15 -> {K0,K1} = {w1, 1-w1} for row M=lane;
        //                lanes 16-31 -> {K2,K3} = zeros.
        float wa = w1[m0 + col];         // in-range read for all lanes
        v2f A;
        A.x = lo ? wa          : 0.0f;
        A.y = lo ? 1.0f - wa   : 0.0f;

        // B (4x16, KxN): lanes 0-15 -> {K0,K1} = {amps_s[n], amps_prev[n]};
        //                lanes 16-31 -> {K2,K3} = zeros.  (c0+col < 112 always)
        float bs = a_cur[c0 + col];
        float bp = a_prv[c0 + col];
        v2f B;
        B.x = lo ? bs : 0.0f;
        B.y = lo ? bp : 0.0f;

        v8f C = {};
        // D = A x B + 0  (EXEC all-ones here: no divergent flow around call)
        v8f D = __builtin_amdgcn_wmma_f32_16x16x4_f32(
            /*neg_a=*/false, A, /*neg_b=*/false, B,
            /*c_mod=*/(short)0, C, /*reuse_a=*/false, /*reuse_b=*/false);

        // D layout: VGPR r -> M=r (lanes 0-15), M=8+r (lanes 16-31); N = col.
        float* op = out + ob_base + m0 * NC + c0;
        const int rowoff = lo ? 0 : 8;
        if (c0 + col < NC) {             // mask 12 dead channels in last tile
            #pragma unroll
            for (int r = 0; r < 8; ++r) {
                op[(r + rowoff) * NC + col] = D[r];
            }
        }
    }
}

extern "C" void kernel_launch(void* const* d_in, const int* in_sizes, int n_in,
                              void* d_out, int out_size, void* d_ws, size_t ws_size,
                              hipStream_t stream) {
    const float* amp = (const float*)d_in[0];   // [8,250,100]
    const float* frq = (const float*)d_in[1];   // [8,250,100]
    float* out      = (float*)d_out;            // 51.4M floats
    float* amps_ws  = (float*)d_ws;             // 200K floats scratch
    float* freq_out = out + NB * NS * NC;       // second tuple output

    const int n_ctrl = NB * NF * NC;            // 200,000
    controls_kernel<<<(n_ctrl + 255) / 256, 256, 0, stream>>>(amp, frq, amps_ws, freq_out);

    upsample_wmma_kernel<<<NB * NF, 256, 0, stream>>>(amps_ws, out);
}